// Blur_35175782154319
// MI455X (gfx1250) — compile-verified
//
#include <hip/hip_runtime.h>
#include <hip/hip_bf16.h>
#include <stdint.h>

// ---------------------------------------------------------------------------
// upfirdn2d blur, pad=(2,1), separable kernel g = [1,3,3,1]/8 * 2 per axis
//   out[n] = 0.25*in[n+1] + 0.75*in[n] + 0.75*in[n-1] + 0.25*in[n-2]
// Implemented as band-matrix WMMAs:  out = V(16x20) * X(20x20) * H(20x16),
// K split into 5 chunks of 4 for V_WMMA_F32_16X16X4_F32.
// Bulk data movement via the Tensor Data Mover (TENSOR_LOAD_TO_LDS).
// Memory-bound workload: 268 MB total traffic -> ~11.5 us floor at 23.3 TB/s.
// ---------------------------------------------------------------------------

typedef __attribute__((ext_vector_type(2))) float        v2f;
typedef __attribute__((ext_vector_type(8))) float        v8f;
typedef __attribute__((ext_vector_type(4))) unsigned int v4u;
typedef __attribute__((ext_vector_type(4))) int          v4i;
typedef __attribute__((ext_vector_type(8))) int          v8i;

#define LDS_S 68   // 68-float row stride: 64-elt rows + 4-DWORD TDM pad

#if defined(__has_builtin)
#if __has_builtin(__builtin_amdgcn_tensor_load_to_lds)
#define HAVE_TDM 1
#endif
#endif

__device__ __forceinline__ float band_w(int i) {
    // w[0..3] = {0.25, 0.75, 0.75, 0.25}; 0 outside the band
    float v = 0.0f;
    if (i == 0 || i == 3) v = 0.25f;
    if (i == 1 || i == 2) v = 0.75f;
    return v;
}

__global__ __launch_bounds__(128)
void Blur_35175782154319_kernel(const float* __restrict__ x,
                                float* __restrict__ out) {
    __shared__ float XT[68][LDS_S];   // input + halo: rows/cols -2..65
    __shared__ float TT[68][LDS_S];   // horizontal-pass intermediate

    const int tid  = threadIdx.x;
    const int lane = tid & 31;
    const int wave = tid >> 5;
    const int n16  = lane & 15;              // row-of-A / col-of-B,C,D per lane
    const int kb   = (lane >> 4) << 1;       // K base: 0 (lanes 0-15) / 2 (16-31)

    const size_t img = blockIdx.x;           // one (n,c) 64x64 image per block
    const float* src = x   + img * 4096;
    float*       dst = out + img * 4096;

    // ---- zero the halo border the TDM tile never writes -------------------
    for (int i = tid; i < 2 * LDS_S; i += 128)        // halo rows 0..1
        (&XT[0][0])[i] = 0.0f;
    for (int i = tid; i < 66 * 4; i += 128) {         // halo cols 0,1,66,67
        int r = 2 + (i >> 2);
        int c = i & 3;
        c = (c < 2) ? c : c + 64;
        XT[r][c] = 0.0f;
    }

    // ---- bulk load: TDM tensor DMA global -> LDS --------------------------
#ifdef HAVE_TDM
    if (wave == 0) {
        const uint64_t ga = (uint64_t)(uintptr_t)src;       // tile start
        const uint32_t la = (uint32_t)(uintptr_t)&XT[2][2]; // LDS dest (low 32b = LDS offset)
        v4u g0;
        g0.x = 1u;                                   // count=1, user descriptor
        g0.y = la;                                   // lds_addr
        g0.z = (uint32_t)ga;                         // global_addr[31:0]
        g0.w = (uint32_t)((ga >> 32) & 0x01FFFFFFu)  // global_addr[56:32]
             | (2u << 30);                           // type=2 ("image")
        v8i g1;
        g1[0] = (int)((2u << 16)        // data_size = 4 bytes
                    | (1u << 20)        // pad_enable
                    | (5u << 22)        // pad_interval: every 64 DWORDs
                    | (3u << 25));      // pad_amount: 4 DWORDs -> stride 68
        g1[1] = (int)(64u << 16);       // tensor_dim0[15:0] = 64
        g1[2] = (int)(64u << 16);       // tensor_dim1[15:0] = 64
        g1[3] = (int)(64u << 16);       // tile_dim0 = 64
        g1[4] = 66;                     // tile_dim1 = 66 (rows 64,65 OOB -> zeros)
        g1[5] = 64;                     // tensor_dim0_stride = 64 elements
        g1[6] = 0;
        g1[7] = 0;
        v4i gz4 = {0, 0, 0, 0};         // 2-D tensor: groups 2/3 unused
        v8i gz8 = {0, 0, 0, 0, 0, 0, 0, 0};
        __builtin_amdgcn_tensor_load_to_lds(g0, g1, gz4, gz4, gz8, 0);
        __builtin_amdgcn_s_wait_tensorcnt(0);
    }
#else
    for (int i = tid; i < 66 * 64; i += 128) {
        int r = i >> 6, c = i & 63;
        XT[2 + r][2 + c] = (r < 64) ? src[r * 64 + c] : 0.0f;
    }
#endif
    __syncthreads();

    // ---- constant band-matrix chunks (identical per-lane values serve both
    //      the B operand of pass 1 and the A operand of pass 2) -------------
    v2f wband[5];
#pragma unroll
    for (int c = 0; c < 5; ++c) {
        int i0 = 4 * c + kb - n16;
        wband[c].x = band_w(i0);
        wband[c].y = band_w(i0 + 1);
    }

    // ---- pass 1: horizontal FIR, T(16x16) = X(16x20) * H(20x16) -----------
    // 5 row-chunks x 4 col-chunks = 20 tiles over the 68 halo rows
    const int rbs[5] = {0, 16, 32, 48, 52};   // last overlaps: benign recompute
#pragma unroll
    for (int t = 0; t < 5; ++t) {
        int tile = wave + 4 * t;              // 0..19
        int rb = rbs[tile % 5];
        int C0 = (tile / 5) * 16;
        v8f acc = {};
#pragma unroll
        for (int c = 0; c < 5; ++c) {
            v2f a = *(const v2f*)&XT[rb + n16][C0 + 4 * c + kb];  // 8B aligned
            acc = __builtin_amdgcn_wmma_f32_16x16x4_f32(
                false, a, false, wband[c], (short)0, acc, false, false);
        }
        int sr = rb + ((lane >> 4) << 3);     // D: VGPR g -> row g / g+8
#pragma unroll
        for (int g = 0; g < 8; ++g)
            TT[sr + g][C0 + n16] = acc[g];
    }
    __syncthreads();

    // ---- pass 2: vertical FIR, out(16x16) = V(16x20) * T(20x16) -----------
#pragma unroll
    for (int t = 0; t < 4; ++t) {
        int tile = wave + 4 * t;              // 0..15
        int R0 = (tile >> 2) * 16;
        int C0 = (tile & 3) * 16;
        v8f acc = {};
#pragma unroll
        for (int c = 0; c < 5; ++c) {
            int r = R0 + 4 * c + kb;          // halo-row index into TT
            v2f b;
            b.x = TT[r][C0 + n16];
            b.y = TT[r + 1][C0 + n16];
            acc = __builtin_amdgcn_wmma_f32_16x16x4_f32(
                false, wband[c], false, b, (short)0, acc, false, false);
        }
        int sr = R0 + ((lane >> 4) << 3);
#pragma unroll
        for (int g = 0; g < 8; ++g)
            dst[(size_t)(sr + g) * 64 + (C0 + n16)] = acc[g];
    }
}

extern "C" void kernel_launch(void* const* d_in, const int* in_sizes, int n_in,
                              void* d_out, int out_size, void* d_ws, size_t ws_size,
                              hipStream_t stream) {
    const float* x = (const float*)d_in[0];
    // d_in[1] is the 4x4 kernel; it is the fixed outer([1,3,3,1]) * 4 / 64,
    // whose exact separable factors [0.25,0.75,0.75,0.25] are baked into the
    // band matrices above (exactly representable in f32).
    float* out = (float*)d_out;
    int n_img = in_sizes[0] / 4096;           // (N*C) 64x64 images = 8192
    Blur_35175782154319_kernel<<<dim3(n_img), dim3(128), 0, stream>>>(x, out);
}